// Center_of_mass_initial_pairwise_46308337385688
// MI455X (gfx1250) — compile-verified
//
#include <hip/hip_runtime.h>

typedef __attribute__((ext_vector_type(2))) float v2f;
typedef __attribute__((ext_vector_type(8))) float v8f;

#define DHW 192
#define VOXELS (192 * 192 * 192)
#define ROWS (192 * 192)        // rows per volume (d*H + h), each 192 wide
#define TILES (ROWS / 16)       // 2304 tiles of 16 rows
#define LDS_STRIDE 196          // 192 + 4 pad: bank-conflict-free for both lane halves
#define INV191 (2.0f / 191.0f)

// ---------------------------------------------------------------- init
__global__ void __launch_bounds__(128) init_ws(float* __restrict__ ws) {
    if (threadIdx.x < 16) ws[threadIdx.x] = 0.0f;
}

// ---------------------------------------------------------------- pass 1
// ws[vol*4 + 0..3] = { S, Sum*gx, Sum*gy, Sum*gz } for vol in {x=0, y=1}
__global__ void __launch_bounds__(128) com_reduce_wmma(const float* __restrict__ xv,
                                                       const float* __restrict__ yv,
                                                       float* __restrict__ ws) {
    __shared__ float lds[4][16 * LDS_STRIDE];
    __shared__ float blk[4];

    const int lane = threadIdx.x & 31;
    const int wave = threadIdx.x >> 5;
    const int vol  = blockIdx.y;
    const float* __restrict__ src_base = vol ? yv : xv;

    const int tile = blockIdx.x * 4 + wave;   // one 16-row tile per wave, exact cover
    const int row0 = tile * 16;

    if (threadIdx.x < 4) blk[threadIdx.x] = 0.0f;

    // Stage 16 rows x 192 f32 (one contiguous 12 KB span) into padded LDS.
    // Fully coalesced: 24 x (global_load_b128 + ds_store_b128) per lane.
    const float* __restrict__ src = src_base + (size_t)row0 * DHW;
    float* dst = &lds[wave][0];
    #pragma unroll
    for (int i = 0; i < 24; ++i) {
        const int e = (i * 32 + lane) * 4;            // stays inside one 192-wide row
        const float4 v = *(const float4*)(src + e);
        const int r = e / DHW, ccol = e % DHW;
        *(float4*)(dst + r * LDS_STRIDE + ccol) = v;  // 16B aligned (196*4 % 16 == 0)
    }
    __syncthreads();

    // WMMA accumulation: D[m][0] = sum_w tile[m][w], D[m][1] = sum_w tile[m][w]*gx[w]
    const int n     = lane & 15;           // A row index AND B/D column index
    const int khalf = (lane >> 4) << 1;    // this half's K offset {0,2}
    const float* __restrict__ arow = &lds[wave][n * LDS_STRIDE];

    // B[k][n] = scale_n * k + bias_n  (col 0: ones, col 1: gx[k], cols >=2: zero)
    const float scale = (n == 1) ? INV191 : 0.0f;
    const float bias  = (n == 0) ? 1.0f : ((n == 1) ? -1.0f : 0.0f);
    float kf = (float)khalf;               // tracks k = w0 + khalf as a float

    v8f c = {};
    #pragma unroll 8
    for (int w0 = 0; w0 < DHW; w0 += 4) {
        v2f a = *(const v2f*)(arow + w0 + khalf);   // A[m][khalf], A[m][khalf+1]
        v2f b;
        b.x = fmaf(scale, kf, bias);
        b.y = b.x + scale;
        c = __builtin_amdgcn_wmma_f32_16x16x4_f32(false, a, false, b, (short)0, c,
                                                  false, false);
        kf += 4.0f;
    }

    // c[j] = D[j + 8*(lane>>4)][n]. Column 0 carries row sums, column 1 carries
    // gx-weighted row sums; columns >= 2 are exactly zero. Fold per-row gy/gz here.
    const int mbase = row0 + ((lane >> 4) << 3);
    float s_p = 0.0f, s_y = 0.0f, s_z = 0.0f;
    #pragma unroll
    for (int j = 0; j < 8; ++j) {
        const int row    = mbase + j;       // row = d*192 + h
        const int hcoord = row % DHW;
        const int dcoord = row / DHW;
        const float gy = (float)hcoord * INV191 - 1.0f;
        const float gz = (float)dcoord * INV191 - 1.0f;
        const float val = c[j];
        s_p += val;
        s_y += val * gy;
        s_z += val * gz;
    }

    if (n == 0) {                 // lanes 0,16: row-sum column -> S, Sy, Sz
        atomicAdd(&blk[0], s_p);
        atomicAdd(&blk[2], s_y);
        atomicAdd(&blk[3], s_z);
    } else if (n == 1) {          // lanes 1,17: gx-weighted column -> Sx
        atomicAdd(&blk[1], s_p);
    }
    __syncthreads();
    if (threadIdx.x < 4) atomicAdd(&ws[vol * 4 + threadIdx.x], blk[threadIdx.x]);
}

// ---------------------------------------------------------------- pass 1b
__global__ void finalize_t(float* __restrict__ ws) {
    if (threadIdx.x == 0) {
        const float sx = ws[0], sy = ws[4];
        ws[8]  = ws[1] / sx - ws[5] / sy;   // t = cm_x - cm_y
        ws[9]  = ws[2] / sx - ws[6] / sy;
        ws[10] = ws[3] / sx - ws[7] / sy;
    }
}

// ---------------------------------------------------------------- pass 2
// out[0 .. V)        : trilinear resample of x at (base + t)
// out[V .. 4V)       : grid = base + t, interleaved (gx, gy, gz)
__global__ void __launch_bounds__(256) sample_kernel(const float* __restrict__ x,
                                                     const float* __restrict__ ws,
                                                     float* __restrict__ out) {
    const int idx = blockIdx.x * 256 + threadIdx.x;      // grid sized exactly
    const float tx = ws[8], ty = ws[9], tz = ws[10];     // uniform -> scalar loads

    const int w  = idx % DHW;
    const int t1 = idx / DHW;
    const int h  = t1 % DHW;
    const int d  = t1 / DHW;

    const float g0 = (float)w * INV191 - 1.0f + tx;
    const float g1 = (float)h * INV191 - 1.0f + ty;
    const float g2 = (float)d * INV191 - 1.0f + tz;

    float* __restrict__ gout = out + (size_t)VOXELS + (size_t)idx * 3;
    gout[0] = g0; gout[1] = g1; gout[2] = g2;

    const float ix = (g0 + 1.0f) * 0.5f * 191.0f;
    const float iy = (g1 + 1.0f) * 0.5f * 191.0f;
    const float iz = (g2 + 1.0f) * 0.5f * 191.0f;
    const float fx = floorf(ix), fy = floorf(iy), fz = floorf(iz);
    const float wx = ix - fx, wy = iy - fy, wz = iz - fz;
    const int ix0 = (int)fx, iy0 = (int)fy, iz0 = (int)fz;
    const int ix1 = ix0 + 1, iy1 = iy0 + 1, iz1 = iz0 + 1;

    auto corner = [&](int zi, int yi, int xi, float wgt) -> float {
        const bool valid = ((unsigned)zi < 192u) & ((unsigned)yi < 192u) & ((unsigned)xi < 192u);
        const int zc = min(max(zi, 0), 191);
        const int yc = min(max(yi, 0), 191);
        const int xc = min(max(xi, 0), 191);
        const float v = x[((size_t)zc * DHW + yc) * DHW + xc];
        return valid ? v * wgt : 0.0f;
    };

    const float owx = 1.0f - wx, owy = 1.0f - wy, owz = 1.0f - wz;
    float r = corner(iz0, iy0, ix0, owz * owy * owx)
            + corner(iz0, iy0, ix1, owz * owy * wx)
            + corner(iz0, iy1, ix0, owz * wy  * owx)
            + corner(iz0, iy1, ix1, owz * wy  * wx)
            + corner(iz1, iy0, ix0, wz  * owy * owx)
            + corner(iz1, iy0, ix1, wz  * owy * wx)
            + corner(iz1, iy1, ix0, wz  * wy  * owx)
            + corner(iz1, iy1, ix1, wz  * wy  * wx);
    out[idx] = r;
}

// ---------------------------------------------------------------- launch
extern "C" void kernel_launch(void* const* d_in, const int* in_sizes, int n_in,
                              void* d_out, int out_size, void* d_ws, size_t ws_size,
                              hipStream_t stream) {
    const float* x = (const float*)d_in[0];
    const float* y = (const float*)d_in[1];
    float* out = (float*)d_out;
    float* ws  = (float*)d_ws;

    init_ws<<<1, 128, 0, stream>>>(ws);

    dim3 gred(TILES / 4, 2);                       // 576 blocks x {x,y}
    com_reduce_wmma<<<gred, 128, 0, stream>>>(x, y, ws);

    finalize_t<<<1, 32, 0, stream>>>(ws);

    sample_kernel<<<VOXELS / 256, 256, 0, stream>>>(x, ws, out);
}